// TopKClassificationLoss_64544768524967
// MI455X (gfx1250) — compile-verified
//
#include <hip/hip_runtime.h>
#include <hip/hip_bf16.h>

#define N_CLASSES 1000
#define NPAD      1024
#define TOPK      5
#define THREADS   256

typedef __attribute__((address_space(1))) int gint_t;   // global i32
typedef __attribute__((address_space(3))) int lint_t;   // LDS i32

// ---------------------------------------------------------------------------
// Kernel 1: one workgroup per batch row.
//   - async-stage row into LDS (CDNA5 GLOBAL_LOAD_ASYNC_TO_LDS path)
//   - parallel count of target's descending-sort position (stable ties)
//   - bitonic sort (descending) of 1024 padded elements in LDS
//   - thread 0: O(n) PAV isotonic regression (decreasing) on y_i = s_i-(n-i),
//     locate block containing pos, emit relu(996 - (v_t - sol[pos]))
// ---------------------------------------------------------------------------
__global__ __launch_bounds__(THREADS)
void topk_loss_row_kernel(const float* __restrict__ x,
                          const int* __restrict__ target,
                          float* __restrict__ row_loss) {
    __shared__ float sh[NPAD];
    __shared__ float blkSum[N_CLASSES];
    __shared__ int   blkCnt[N_CLASSES];
    __shared__ float blkMean[N_CLASSES];
    __shared__ int   sh_pos;

    const int row = blockIdx.x;
    const int tid = threadIdx.x;
    const float* __restrict__ src = x + (size_t)row * N_CLASSES;

    if (tid == 0) sh_pos = 0;

    // ---- stage global -> LDS ----
#if defined(__AMDGCN__) && __has_builtin(__builtin_amdgcn_global_load_async_to_lds_b32)
    for (int i = tid; i < N_CLASSES; i += THREADS) {
        __builtin_amdgcn_global_load_async_to_lds_b32(
            (gint_t*)(src + i),
            (lint_t*)&sh[i],
            /*offset=*/0, /*cpol=*/0);
    }
#if __has_builtin(__builtin_amdgcn_s_wait_asynccnt)
    __builtin_amdgcn_s_wait_asynccnt(0);
#else
    asm volatile("s_wait_asynccnt 0" ::: "memory");
#endif
#else
    for (int i = tid; i < N_CLASSES; i += THREADS) sh[i] = src[i];
#endif
    for (int i = N_CLASSES + tid; i < NPAD; i += THREADS)
        sh[i] = -__builtin_inff();   // pad sorts to the tail (descending)
    __syncthreads();

    // ---- target position in stable descending sort ----
    const int   t  = target[row];
    const float vt = sh[t];
    {
        int c = 0;
        for (int j = tid; j < N_CLASSES; j += THREADS) {
            float v = sh[j];
            c += (v > vt) || ((v == vt) && (j < t));
        }
        atomicAdd(&sh_pos, c);   // integer LDS atomic: deterministic
    }
    __syncthreads();

    // ---- bitonic sort, descending, NPAD elements, THREADS threads ----
    for (int k = 2; k <= NPAD; k <<= 1) {
        for (int j = k >> 1; j > 0; j >>= 1) {
            for (int i = tid; i < NPAD; i += THREADS) {
                int ixj = i ^ j;
                if (ixj > i) {
                    float a = sh[i], b = sh[ixj];
                    bool up   = ((i & k) == 0);          // descending run
                    bool swap = up ? (a < b) : (a > b);
                    if (swap) { sh[i] = b; sh[ixj] = a; }
                }
            }
            __syncthreads();
        }
    }

    // ---- PAV isotonic regression (nonincreasing) + loss ----
    if (tid == 0) {
        int nb = 0;
        for (int i = 0; i < N_CLASSES; ++i) {
            float sum = sh[i] - (float)(N_CLASSES - i);  // y_i = s_i - w_i
            int   c   = 1;
            float m   = sum;
            // merge while previous block mean < current block mean
            while (nb > 0 && blkMean[nb - 1] < m) {
                sum += blkSum[nb - 1];
                c   += blkCnt[nb - 1];
                m    = sum / (float)c;
                --nb;
            }
            blkSum[nb]  = sum;
            blkCnt[nb]  = c;
            blkMean[nb] = m;
            ++nb;
        }
        // locate block containing pos; sol[pos] = that block's mean
        int   pos    = sh_pos;
        int   acc    = 0;
        float solpos = 0.0f;
        for (int bb = 0; bb < nb; ++bb) {
            acc += blkCnt[bb];
            if (pos < acc) { solpos = blkMean[bb]; break; }
        }
        float rank = vt - solpos;
        float li   = (float)(N_CLASSES - TOPK + 1) - rank;  // 996 - rank
        row_loss[row] = (li > 0.0f) ? li : 0.0f;
    }
}

// ---------------------------------------------------------------------------
// Kernel 2: deterministic fixed-order mean of the 64 per-row losses.
// ---------------------------------------------------------------------------
__global__ void topk_loss_reduce_kernel(const float* __restrict__ row_loss,
                                        float* __restrict__ out, int nrows) {
    if (blockIdx.x == 0 && threadIdx.x == 0) {
        float s = 0.0f;
        for (int i = 0; i < nrows; ++i) s += row_loss[i];
        out[0] = s / (float)nrows;
    }
}

extern "C" void kernel_launch(void* const* d_in, const int* in_sizes, int n_in,
                              void* d_out, int out_size, void* d_ws, size_t ws_size,
                              hipStream_t stream) {
    (void)n_in; (void)out_size; (void)ws_size;
    const float* x      = (const float*)d_in[0];   // [B, 1000] fp32
    const int*   target = (const int*)d_in[1];     // [B] int32
    const int    B      = in_sizes[1];             // 64

    float* row_loss = (float*)d_ws;                // B floats of scratch

    topk_loss_row_kernel<<<B, THREADS, 0, stream>>>(x, target, row_loss);
    topk_loss_reduce_kernel<<<1, 32, 0, stream>>>(row_loss, (float*)d_out, B);
}